// NucleicAcidFeature_29703993819767
// MI455X (gfx1250) — compile-verified
//
#include <hip/hip_runtime.h>

// ---------------------------------------------------------------------------
// Problem constants (match the JAX reference)
// ---------------------------------------------------------------------------
static constexpr int kB = 8;      // graphs
static constexpr int kL = 512;    // nodes per graph
static constexpr int kC = 6;      // real atom channels
static constexpr int kCP = 8;     // channels padded to 8 (2x2 node pairs / 16x16 WMMA tile)
static constexpr int kK = 9;      // knn
static constexpr int kN = kB * kL;            // 4096 nodes total
static constexpr int kSlots = kN * kCP;       // 32768 padded channel slots
#define BIGF 1.0e10f
#define BIG2 1.0e20f                           // squared-domain mask addend

typedef __attribute__((ext_vector_type(2))) float v2f;
typedef __attribute__((ext_vector_type(8))) float v8f;

// Output layout (floats, concatenated in reference return order)
static constexpr int OFF_DKNN = 0;
static constexpr int OFF_SRC  = kN * kK;          // 36864
static constexpr int OFF_DST  = 2 * kN * kK;      // 73728
static constexpr int OFF_VAL  = 3 * kN * kK;      // 110592
static constexpr int GLBLEN   = kB * (2 * kL - 1);    // 8184
static constexpr int SEQLEN   = kB * 2 * (kL - 2);    // 8160
static constexpr int OFF_GLB  = 4 * kN * kK;          // 147456
static constexpr int OFF_SEQ  = OFF_GLB + 2 * GLBLEN; // 163824

// Async global->LDS copy of 16 bytes per lane (CDNA5 ASYNCcnt path).
// Generic->LDS address: ISA 10.2 — low 32 bits of the address (truncation).
__device__ __forceinline__ void async_copy_b128(const void* g, void* l) {
  asm volatile("global_load_async_to_lds_b128 %0, %1, off"
               :: "v"((unsigned int)(size_t)l), "v"(g) : "memory");
}
__device__ __forceinline__ void wait_async0() {
  asm volatile("s_wait_asynccnt 0" ::: "memory");
}

// VALU-only xor-permute min within aligned 8-lane groups via DPP8
// (replaces ds_bpermute round-trips in the hot loop).
// DPP8 selector: lane i reads lane sel[i] of its 8-lane row; 3 bits per lane.
#define DPP8_XOR1 (1|(0<<3)|(3<<6)|(2<<9)|(5<<12)|(4<<15)|(7<<18)|(6<<21))
#define DPP8_XOR2 (2|(3<<3)|(0<<6)|(1<<9)|(6<<12)|(7<<15)|(4<<18)|(5<<21))
#define DPP8_XOR4 (4|(5<<3)|(6<<6)|(7<<9)|(0<<12)|(1<<15)|(2<<18)|(3<<21))

template <int SEL>
__device__ __forceinline__ float min_dpp8(float v) {
  float p = __int_as_float(__builtin_amdgcn_mov_dpp8(__float_as_int(v), SEL));
  return fminf(v, p);
}

// ---------------------------------------------------------------------------
// Kernel 1: build padded WMMA-friendly layout in workspace.
//   Xp   : float4[kSlots]  (x,y,z,0); zeros for pad channels c>=6
//   sq   : float [kSlots]  |x|^2 per channel slot
//   padf : float [kSlots]  mask flag: (AP==0 or c>=6) + node-is-BOS(S==0)
//          (BOS folded in: any pad/BOS involvement makes the flag-sum > 0)
// ---------------------------------------------------------------------------
__global__ __launch_bounds__(256) void nuc_prep(
    const float* __restrict__ X, const int* __restrict__ AP,
    const int* __restrict__ S, float4* __restrict__ Xp,
    float* __restrict__ sq, float* __restrict__ padf)
{
  int gid = blockIdx.x * blockDim.x + threadIdx.x;
  if (gid >= kSlots) return;
  int n = gid >> 3;     // global node index
  int c = gid & 7;      // padded channel index
  float x = 0.f, y = 0.f, z = 0.f;
  float pad = 1.0f;
  if (c < kC) {
    int base = (n * kC + c) * 3;
    x = X[base + 0]; y = X[base + 1]; z = X[base + 2];
    pad = (AP[n * kC + c] == 0) ? 1.0f : 0.0f;
  }
  if (S[n] == 0) pad += 1.0f;               // BOS/global node: mask entire node
  Xp[gid]   = make_float4(x, y, z, 0.0f);
  sq[gid]   = x * x + y * y + z * z;
  padf[gid] = pad;
}

// ---------------------------------------------------------------------------
// Kernel 2: async-stage column panel to LDS -> WMMA Gram tiles (squared dist)
//           -> per-node-pair min (DPP8) -> top-9 (sqrt only at output).
// Grid: kB * (kL/16) blocks of 256 threads (8 waves).
// Wave w owns row nodes {chunk*16 + 2w, +1} (16 channel-rows = one A operand).
// ---------------------------------------------------------------------------
__global__ __launch_bounds__(256) void nuc_knn(
    const float4* __restrict__ Xp, const float* __restrict__ sq,
    const float* __restrict__ padf, const int* __restrict__ sec_pos,
    float* __restrict__ out)
{
  // LDS: 16KB col coords + 4KB col sq + 4KB col mask + 32KB distance rows = 56KB
  __shared__ float4 sXp[1024];        // 128 col nodes * 8 channels
  __shared__ float  sSq[1024];
  __shared__ float  sPd[1024];
  __shared__ float  rowdist[16][kL];  // squared min distance per node pair

  const int b     = blockIdx.x >> 5;        // graph
  const int chunk = blockIdx.x & 31;        // 16-node chunk within graph
  const int wave  = threadIdx.x >> 5;
  const int lane  = threadIdx.x & 31;
  const int hi    = lane >> 4;              // lane half: K{0,1}/rows 0-7 vs K{2,3}/rows 8-15
  const int l15   = lane & 15;

  const int nodeBase    = chunk * 16 + wave * 2;            // first row node (in-graph)
  const int rowChanBase = b * (kL * kCP) + nodeBase * kCP;  // channel-slot base

  // A operand (16x4 f32): lane l holds M=l&15, VGPR0/1 = K{0,1} (lanes<16) or K{2,3}.
  float4 xa = Xp[rowChanBase + l15];
  v2f av; av[0] = hi ? xa.z : xa.x; av[1] = hi ? xa.w : xa.y;

  // Loop-invariant per-lane row stats: sq / mask for the 8 channel rows this
  // lane-half sees in the C/D tile (M = hi*8 + r). Mask already includes BOS.
  const float* sqp = sq + rowChanBase + hi * 8;
  float srr[8];
  { float4 t0 = *(const float4*)sqp; float4 t1 = *(const float4*)(sqp + 4);
    srr[0]=t0.x; srr[1]=t0.y; srr[2]=t0.z; srr[3]=t0.w;
    srr[4]=t1.x; srr[5]=t1.y; srr[6]=t1.z; srr[7]=t1.w; }
  const float* pdp = padf + rowChanBase + hi * 8;
  float prr[8];
  { float4 t0 = *(const float4*)pdp; float4 t1 = *(const float4*)(pdp + 4);
    prr[0]=t0.x; prr[1]=t0.y; prr[2]=t0.z; prr[3]=t0.w;
    prr[4]=t1.x; prr[5]=t1.y; prr[6]=t1.z; prr[7]=t1.w; }

  // Sweep 4 chunks of 128 column nodes; each chunk async-staged to LDS.
  for (int ct = 0; ct < 4; ++ct) {
    __syncthreads();                        // previous chunk fully consumed
    {
      const float4* gXp = Xp   + b * (kL * kCP) + ct * 1024;
      const float*  gSq = sq   + b * (kL * kCP) + ct * 1024;
      const float*  gPd = padf + b * (kL * kCP) + ct * 1024;
#pragma unroll
      for (int i = 0; i < 4; ++i) {
        int idx = threadIdx.x + 256 * i;
        async_copy_b128(gXp + idx, &sXp[idx]);
      }
      int f4 = threadIdx.x * 4;             // 4 floats (16B) per thread
      async_copy_b128(gSq + f4, &sSq[f4]);
      async_copy_b128(gPd + f4, &sPd[f4]);
      wait_async0();                        // this wave's copies landed in LDS
    }
    __syncthreads();                        // everyone's copies visible

#pragma unroll 2
    for (int jtl = 0; jtl < 64; ++jtl) {    // 2 col nodes per 16x16 tile
      const int ls = jtl * 16 + l15;
      float4 xb = sXp[ls];
      v2f bv; bv[0] = hi ? xb.z : xb.x; bv[1] = hi ? xb.w : xb.y;
      const float sc = sSq[ls];
      const float pc = sPd[ls];             // col mask (pad + BOS)

      // Gram tile: dot[M,N] = x_row(M) . x_col(N)  (K=4: x,y,z,0)
      v8f acc = {};
      acc = __builtin_amdgcn_wmma_f32_16x16x4_f32(
          false, av, false, bv, (short)0, acc, false, false);

      float m2 = 3.0e38f;
#pragma unroll
      for (int r = 0; r < 8; ++r) {
        float d2 = srr[r] + sc - 2.0f * acc[r];       // squared distance
        d2 += (prr[r] + pc > 0.0f) ? BIG2 : 0.0f;     // masked -> +1e20
        m2 = fminf(m2, d2);
      }
      // Min across the 8 lanes sharing one (rowNode, colNode) 8x8 sub-block:
      // pure-VALU DPP8 xor-permutes (no LDS round-trips).
      m2 = min_dpp8<DPP8_XOR1>(m2);
      m2 = min_dpp8<DPP8_XOR2>(m2);
      m2 = min_dpp8<DPP8_XOR4>(m2);
      if ((lane & 7) == 0)
        rowdist[wave * 2 + hi][ct * 128 + jtl * 2 + (l15 >> 3)] = m2;
    }
  }
  __syncthreads();

  // Top-9 smallest per row (squared domain; sqrt monotonic => same order),
  // JAX top_k tie-breaking (equal values -> lower index).
  const int OFFD = OFF_DKNN, OFFS = OFF_SRC, OFFT = OFF_DST, OFFV = OFF_VAL;
  for (int rr = 0; rr < 2; ++rr) {
    const int gi = b * kL + nodeBase + rr;          // global row node
    float* row = rowdist[wave * 2 + rr];
    const int secRow = sec_pos[gi];
    for (int t = 0; t < kK; ++t) {
      float bvv = 3.4e38f; int bi = 0x7fffffff;
      for (int j = lane; j < kL; j += 32) {
        float v = row[j];
        if (v < bvv) { bvv = v; bi = j; }           // lane-local: first (lowest j) wins ties
      }
#pragma unroll
      for (int s = 16; s; s >>= 1) {                // wave argmin with index tie-break
        float ov = __shfl_xor(bvv, s);
        int   oi = __shfl_xor(bi, s);
        if (ov < bvv || (ov == bvv && oi < bi)) { bvv = ov; bi = oi; }
      }
      if (lane == 0) {
        const int gj = b * kL + bi;
        const bool val = (bvv < BIG2) && (sec_pos[gj] == secRow);
        out[OFFD + gi * kK + t] = (bvv < BIG2)
                                ? __builtin_sqrtf(fmaxf(bvv, 0.0f)) : BIGF;
        out[OFFS + gi * kK + t] = (float)gi;
        out[OFFT + gi * kK + t] = val ? (float)gj : -1.0f;
        out[OFFV + gi * kK + t] = val ? 1.0f : 0.0f;
        row[bi] = 3.4e38f;                          // knock out selected entry
      }
      __builtin_amdgcn_wave_barrier();              // keep LDS RAW in program order
    }
  }
}

// ---------------------------------------------------------------------------
// Kernel 3: static global + sequential edge lists (pure index math).
// ---------------------------------------------------------------------------
__global__ __launch_bounds__(256) void nuc_edges(float* __restrict__ out)
{
  int gid = blockIdx.x * blockDim.x + threadIdx.x;
  if (gid < 2 * GLBLEN) {
    const int per = 2 * kL - 1;                      // 1023 per graph
    int rowsel = gid / GLBLEN;
    int rem    = gid % GLBLEN;
    int g = rem / per, p = rem % per, off = g * kL;
    int v;
    if (rowsel == 0) v = (p < kL) ? 0 : (p - kL + 1);   // src: [0]*512, 1..511
    else             v = (p < kL) ? p : 0;              // dst: 0..511, [0]*511
    out[OFF_GLB + gid] = (float)(v + off);
  } else if (gid < 2 * GLBLEN + 2 * SEQLEN) {
    const int per = 2 * (kL - 2);                    // 1020 per graph
    const int half = kL - 2;                         // 510
    int rel    = gid - 2 * GLBLEN;
    int rowsel = rel / SEQLEN;
    int rem    = rel % SEQLEN;
    int g = rem / per, p = rem % per, off = g * kL;
    int v;
    if (rowsel == 0) v = (p < half) ? (1 + p) : (2 + (p - half));  // src: a, a+1 (a=1..510)
    else             v = (p < half) ? (2 + p) : (1 + (p - half));  // dst: a+1, a
    out[OFF_SEQ + rel] = (float)(v + off);
  }
}

// ---------------------------------------------------------------------------
// Launch. Inputs: X(f32), AP(i32), S(i32), sec_pos(i32), batch_id(i32), k(i32).
// Workspace: Xp4 (512KB) | sq (128KB) | mask (128KB) < 1 MB.
// ---------------------------------------------------------------------------
extern "C" void kernel_launch(void* const* d_in, const int* in_sizes, int n_in,
                              void* d_out, int out_size, void* d_ws, size_t ws_size,
                              hipStream_t stream)
{
  const float* X       = (const float*)d_in[0];
  const int*   AP      = (const int*)d_in[1];
  const int*   S       = (const int*)d_in[2];
  const int*   sec_pos = (const int*)d_in[3];
  float* out = (float*)d_out;

  char* ws = (char*)d_ws;
  float4* Xp = (float4*)(ws);
  float*  sq = (float*)(ws + (size_t)kSlots * 16);
  float*  pd = (float*)(ws + (size_t)kSlots * 16 + (size_t)kSlots * 4);

  nuc_prep<<<(kSlots + 255) / 256, 256, 0, stream>>>(X, AP, S, Xp, sq, pd);
  nuc_knn<<<kB * (kL / 16), 256, 0, stream>>>(Xp, sq, pd, sec_pos, out);
  const int nEdgeThreads = 2 * GLBLEN + 2 * SEQLEN;   // 32688
  nuc_edges<<<(nEdgeThreads + 255) / 256, 256, 0, stream>>>(out);
}